// dRNN_27376121545253
// MI455X (gfx1250) — compile-verified
//
#include <hip/hip_runtime.h>
#include <math.h>

typedef __attribute__((ext_vector_type(2))) float v2f;
typedef __attribute__((ext_vector_type(4))) float v4f;
typedef __attribute__((ext_vector_type(8))) float v8f;

#define N_IN   64
#define N_HID  256
#define N_OUT  16
#define BATCH  16
#define TSTEPS 64

#define C_U         0.45f
#define C_SU0       0.45f      /* U_STD + 1 = -0.55 + 1 */
#define C_ALPHA_STD 0.006667f
#define C_ALPHA_STF 0.05f
#define C_DT_S      0.01f      /* DT/1000 */
#define C_DTM       0.1f       /* DT/TAU_MEM */

// ---------------- init: h0 = 0, sx0 = 1, su0 = 0.45, rec_ws = 0 ----------------
__global__ void init_state(float* __restrict__ h_all,
                           float* __restrict__ sx_all,
                           float* __restrict__ su_all,
                           float* __restrict__ rec_ws) {
  const size_t n = (size_t)BATCH * N_HID * N_HID;   // 1,048,576
  const size_t stride = (size_t)gridDim.x * blockDim.x;
  for (size_t i = (size_t)blockIdx.x * blockDim.x + threadIdx.x; i < n; i += stride) {
    sx_all[i] = 1.0f;
    su_all[i] = C_SU0;
    if (i < (size_t)BATCH * N_HID) { h_all[i] = 0.0f; rec_ws[i] = 0.0f; }
  }
}

// ---------------- xin = x(T*B,64) @ W_in(64,256), fp32 WMMA ----------------
__global__ __launch_bounds__(32) void gemm_xin(const float* __restrict__ x,
                                               const float* __restrict__ W_in,
                                               float* __restrict__ xin) {
  const int mt = blockIdx.x;              // 0..63  (16-row tiles over T*B=1024)
  const int nt = blockIdx.y;              // 0..15  (16-col tiles over 256)
  const int lane = threadIdx.x;           // wave32, EXEC all ones
  const int r = lane & 15;
  const int koff = (lane >> 4) << 1;      // 0 or 2 per 16x16x4 A/B layout
  v8f acc = {};
  for (int k = 0; k < N_IN; k += 4) {
    v2f a, bm;
    const float* arow = x + (size_t)(mt * 16 + r) * N_IN + (k + koff);
    a.x = arow[0];
    a.y = arow[1];
    bm.x = W_in[(size_t)(k + koff) * N_HID + nt * 16 + r];
    bm.y = W_in[(size_t)(k + koff + 1) * N_HID + nt * 16 + r];
    acc = __builtin_amdgcn_wmma_f32_16x16x4_f32(false, a, false, bm,
                                                (short)0, acc, false, false);
  }
  const int mbase = mt * 16 + ((lane >> 4) ? 8 : 0);
  const int col = nt * 16 + (lane & 15);
#pragma unroll
  for (int v = 0; v < 8; ++v)
    xin[(size_t)(mbase + v) * N_HID + col] = acc[v];
}

// ---- plasticity update: grid (BATCH, 16 row-blocks), 256 thr (8 waves) ----
// Each lane owns 8 contiguous columns k = 8*lane..8*lane+7  (b128 traffic).
__global__ __launch_bounds__(256) void plast_kernel(const float* __restrict__ W_rec,
                                                    const float* __restrict__ h_all,
                                                    float* __restrict__ sx_all,
                                                    float* __restrict__ su_all,
                                                    float* __restrict__ rec_ws,
                                                    int t) {
  const int b  = blockIdx.x;          // batch element
  const int rb = blockIdx.y;          // 16-row block
  const int tid = threadIdx.x;
  const int wave = tid >> 5;          // 0..7 (wave32)
  const int lane = tid & 31;

  __shared__ float s_rec[8][N_HID];   // per-wave rec partials (8 KB)

  const float* h_old = h_all + ((size_t)t * BATCH + b) * N_HID;
  const size_t matOld = ((size_t)t * BATCH + b) * (size_t)(N_HID * N_HID);
  const size_t matStep = (size_t)BATCH * N_HID * N_HID;
  const float* sx_o = sx_all + matOld;
  const float* su_o = su_all + matOld;
  float* sx_n = sx_all + matOld + matStep;
  float* su_n = su_all + matOld + matStep;

  // h values for this lane's columns, kept in registers (row-invariant)
  const v4f hA = *(const v4f*)(h_old + 8 * lane);
  const v4f hB = *(const v4f*)(h_old + 8 * lane + 4);
  const float hreg[8] = {hA.x, hA.y, hA.z, hA.w, hB.x, hB.y, hB.z, hB.w};

  // sumh = sum_k h[k]   (v = U*(sumh - dot(su_row, h)))
  float sp = 0.f;
#pragma unroll
  for (int u = 0; u < 8; ++u) sp += hreg[u];
#pragma unroll
  for (int off = 16; off > 0; off >>= 1) sp += __shfl_xor(sp, off, 32);
  const float sumh = sp;

  float racc[8];
#pragma unroll
  for (int u = 0; u < 8; ++u) racc[u] = 0.f;

#pragma unroll
  for (int rr = 0; rr < 2; ++rr) {
    const int i = rb * 16 + wave + rr * 8;       // presynaptic row
    const float hi = h_old[i];
    const size_t rowOff = (size_t)i * N_HID + 8 * lane;

    const v4f suA = *(const v4f*)(su_o + rowOff);
    const v4f suB = *(const v4f*)(su_o + rowOff + 4);
    const v4f sxA = *(const v4f*)(sx_o + rowOff);
    const v4f sxB = *(const v4f*)(sx_o + rowOff + 4);
    const v4f wA  = *(const v4f*)(W_rec + rowOff);
    const v4f wB  = *(const v4f*)(W_rec + rowOff + 4);
    const float su_row[8] = {suA.x, suA.y, suA.z, suA.w, suB.x, suB.y, suB.z, suB.w};
    const float sx_row[8] = {sxA.x, sxA.y, sxA.z, sxA.w, sxB.x, sxB.y, sxB.z, sxB.w};
    const float w_row[8]  = {wA.x,  wA.y,  wA.z,  wA.w,  wB.x,  wB.y,  wB.z,  wB.w};

    float p = 0.f;
#pragma unroll
    for (int u = 0; u < 8; ++u) p += su_row[u] * hreg[u];
#pragma unroll
    for (int off = 16; off > 0; off >>= 1) p += __shfl_xor(p, off, 32);
    const float v = C_U * (sumh - p);            // uses OLD su, per reference

    float sxn[8], sun[8];
#pragma unroll
    for (int u = 0; u < 8; ++u) {
      float sxv = sx_row[u] + C_ALPHA_STD * (1.f - sx_row[u])
                  - C_DT_S * su_row[u] * sx_row[u] * hi;
      float suv = su_row[u] + C_ALPHA_STF * (C_U - su_row[u]) + C_DT_S * v;
      sxv = fminf(fmaxf(sxv, 0.f), 1.f);
      suv = fminf(fmaxf(suv, 0.f), 1.f);
      sxn[u] = sxv;
      sun[u] = suv;
      racc[u] += hi * sxv * suv * w_row[u];      // rec[j] partial, j = 8*lane+u
    }
    *(v4f*)(sx_n + rowOff)     = (v4f){sxn[0], sxn[1], sxn[2], sxn[3]};
    *(v4f*)(sx_n + rowOff + 4) = (v4f){sxn[4], sxn[5], sxn[6], sxn[7]};
    *(v4f*)(su_n + rowOff)     = (v4f){sun[0], sun[1], sun[2], sun[3]};
    *(v4f*)(su_n + rowOff + 4) = (v4f){sun[4], sun[5], sun[6], sun[7]};
  }

  // LDS pre-reduction across the 8 waves, then one atomic per column
#pragma unroll
  for (int u = 0; u < 8; ++u) s_rec[wave][8 * lane + u] = racc[u];
  __syncthreads();

  float rec = 0.f;
#pragma unroll
  for (int w = 0; w < 8; ++w) rec += s_rec[w][tid];
  atomicAdd(&rec_ws[(size_t)b * N_HID + tid], rec);
}

// ---- h_new = relu(...) for all batches, then y[t] via fp32 WMMA; resets rec ----
__global__ __launch_bounds__(256) void update_out(const float* __restrict__ W_bias,
                                                  const float* __restrict__ xin,
                                                  const float* __restrict__ W_out,
                                                  const float* __restrict__ W_out_b,
                                                  float* __restrict__ rec_ws,
                                                  float* __restrict__ h_all,
                                                  float* __restrict__ y_seq,
                                                  int t) {
  const int tid = threadIdx.x;
  __shared__ float s_hn[BATCH * N_HID];          // 16 KB
  const float* h_old = h_all + (size_t)t * BATCH * N_HID;
  float* h_new = h_all + (size_t)(t + 1) * BATCH * N_HID;

#pragma unroll
  for (int q = 0; q < BATCH; ++q) {
    const int e = tid + 256 * q;                 // e = b*256 + j
    const int j = e & (N_HID - 1);
    const float pre = (1.f - C_DTM) * h_old[e]
                    + C_DTM * (rec_ws[e] + W_bias[j])
                    + xin[(size_t)t * BATCH * N_HID + e];
    const float hn = fmaxf(pre, 0.f);
    h_new[e] = hn;
    s_hn[e] = hn;
    rec_ws[e] = 0.f;                             // reset for next step's atomics
  }
  __syncthreads();

  if (tid < 32) {                                // wave 0 only, EXEC all ones
    const int lane = tid;
    const int r = lane & 15;
    const int koff = (lane >> 4) << 1;
    v8f acc = {};
    for (int k = 0; k < N_HID; k += 4) {
      v2f a, bm;
      a.x = s_hn[r * N_HID + (k + koff)];
      a.y = s_hn[r * N_HID + (k + koff + 1)];
      bm.x = W_out[(size_t)(k + koff) * N_OUT + r];
      bm.y = W_out[(size_t)(k + koff + 1) * N_OUT + r];
      acc = __builtin_amdgcn_wmma_f32_16x16x4_f32(false, a, false, bm,
                                                  (short)0, acc, false, false);
    }
    const int m0 = (lane >> 4) ? 8 : 0;
    const int n = lane & 15;
#pragma unroll
    for (int v = 0; v < 8; ++v) {
      const float z = acc[v] + W_out_b[n];
      y_seq[(size_t)t * BATCH * N_OUT + (size_t)(m0 + v) * N_OUT + n] =
          1.0f / (1.0f + expf(-z));
    }
  }
}

extern "C" void kernel_launch(void* const* d_in, const int* in_sizes, int n_in,
                              void* d_out, int out_size, void* d_ws, size_t ws_size,
                              hipStream_t stream) {
  (void)in_sizes; (void)n_in; (void)out_size; (void)ws_size;
  const float* x      = (const float*)d_in[0];
  const float* W_in   = (const float*)d_in[1];
  const float* W_rec  = (const float*)d_in[2];
  const float* W_bias = (const float*)d_in[3];
  const float* W_out  = (const float*)d_in[4];
  const float* W_outb = (const float*)d_in[5];

  float* out = (float*)d_out;
  const size_t n_h  = (size_t)(TSTEPS + 1) * BATCH * N_HID;                 // 266,240
  const size_t n_y  = (size_t)TSTEPS * BATCH * N_OUT;                       // 16,384
  const size_t n_sx = (size_t)(TSTEPS + 1) * BATCH * N_HID * N_HID;         // 68,157,440
  float* h_all  = out;
  float* y_seq  = out + n_h;
  float* sx_all = out + n_h + n_y;
  float* su_all = out + n_h + n_y + n_sx;

  float* xin    = (float*)d_ws;                        // (T,B,H): 1 MB
  float* rec_ws = xin + (size_t)TSTEPS * BATCH * N_HID; // (B,H): 16 KB

  init_state<<<1024, 256, 0, stream>>>(h_all, sx_all, su_all, rec_ws);

  dim3 gg((TSTEPS * BATCH) / 16, N_HID / 16);          // 64 x 16 tiles
  gemm_xin<<<gg, 32, 0, stream>>>(x, W_in, xin);

  dim3 gp(BATCH, N_HID / 16);                          // 16 x 16 = 256 WGs/step
  for (int t = 0; t < TSTEPS; ++t) {
    plast_kernel<<<gp, 256, 0, stream>>>(W_rec, h_all, sx_all, su_all, rec_ws, t);
    update_out<<<1, 256, 0, stream>>>(W_bias, xin, W_out, W_outb,
                                      rec_ws, h_all, y_seq, t);
  }
}